// ModelNew_25056839205361
// MI455X (gfx1250) — compile-verified
//
#include <hip/hip_runtime.h>

// diag(A) @ B : out[i][j] = A[i] * B[i][j]
// N = M = 8192, fp32. Pure streaming op (AI = 0.125 FLOP/byte); memory floor
// at 23.3 TB/s is ~22 us for 512 MB moved. Strategy:
//   - one block == one row  -> A[row] is block-uniform (s_load_b32 broadcast)
//   - 8 x b128 NT loads issued back-to-back per thread (max MLP per wave)
//   - NT stores (output written once, no reuse; both streams > 192 MB L2)

typedef float v4f __attribute__((ext_vector_type(4)));

#define COLS   8192u
#define M4     (COLS / 4u)     // 2048 float4 per row
#define TPB    256u
#define UNROLL 8u              // TPB * UNROLL == M4 -> block covers exactly one row
#define ROWS   8192u           // grid = 8192 blocks

__global__ __launch_bounds__(TPB) void diag_scale_row(
    const float* __restrict__ A,
    const v4f*  __restrict__ B,
    v4f*        __restrict__ O)
{
    const unsigned row  = blockIdx.x;                 // SGPR-uniform
    const float    a    = A[row];                     // scalar load, broadcast
    const unsigned base = row * M4 + threadIdx.x;     // element (float4) index

    // Phase 1: issue all 8 b128 NT loads (consecutive lanes -> coalesced;
    // the it*TPB offsets fold into the 24-bit instruction offset, so this is
    // 8 loads off one address register, all in flight before the first wait).
    v4f b[UNROLL];
#pragma unroll
    for (unsigned it = 0; it < UNROLL; ++it) {
        b[it] = __builtin_nontemporal_load(&B[base + it * TPB]);
    }

    // Phase 2: scale (v_pk_mul_f32 x2 per vec) and stream out NT.
#pragma unroll
    for (unsigned it = 0; it < UNROLL; ++it) {
        v4f o;
        o.x = a * b[it].x;
        o.y = a * b[it].y;
        o.z = a * b[it].z;
        o.w = a * b[it].w;
        __builtin_nontemporal_store(o, &O[base + it * TPB]);
    }
}

extern "C" void kernel_launch(void* const* d_in, const int* in_sizes, int n_in,
                              void* d_out, int out_size, void* d_ws, size_t ws_size,
                              hipStream_t stream)
{
    (void)in_sizes; (void)n_in; (void)out_size; (void)d_ws; (void)ws_size;

    const float* A = (const float*)d_in[0];   // 8192 floats
    const v4f*   B = (const v4f*)d_in[1];     // 8192*8192 floats as float4
    v4f*         O = (v4f*)d_out;             // 8192*8192 floats as float4

    diag_scale_row<<<ROWS, TPB, 0, stream>>>(A, B, O);
}